// GCN_20890720928309
// MI455X (gfx1250) — compile-verified
//
#include <hip/hip_runtime.h>

#define N_NODES  100000
#define N_EDGES  1600000
#define N_GRAPHS 512
#define DH       128
#define N_LAYERS 6

typedef __attribute__((ext_vector_type(2))) float v2f;
typedef __attribute__((ext_vector_type(8))) float v8f;

// ---------------- init: deg=1 (self loop), zero pool buffers ----------------
__global__ __launch_bounds__(256) void k_init(float* __restrict__ deg,
                                              float* __restrict__ sums,
                                              float* __restrict__ counts) {
    int i = blockIdx.x * 256 + threadIdx.x;
    if (i < N_NODES)      deg[i]    = 1.0f;
    if (i < N_GRAPHS*DH)  sums[i]   = 0.0f;
    if (i < N_GRAPHS)     counts[i] = 0.0f;
}

__global__ __launch_bounds__(256) void k_deg(const int* __restrict__ dst,
                                             float* __restrict__ deg) {
    int e = blockIdx.x * 256 + threadIdx.x;
    if (e < N_EDGES) atomicAdd(&deg[dst[e]], 1.0f);
}

__global__ __launch_bounds__(256) void k_dinv(float* __restrict__ deg) {
    int i = blockIdx.x * 256 + threadIdx.x;
    if (i < N_NODES) deg[i] = rsqrtf(deg[i]);
}

__global__ __launch_bounds__(256) void k_counts(const int* __restrict__ batch,
                                                float* __restrict__ counts) {
    int n = blockIdx.x * 256 + threadIdx.x;
    if (n < N_NODES) atomicAdd(&counts[batch[n]], 1.0f);
}

// ---------------- GEMM: C[N x 128] = relu?(A[N x 128]) * W[128 x 128] -------
// 256 threads = 8 waves. W staged in LDS (64KB). Wave w computes rows
// [blk*128 + w*16, +16) across all 128 columns: 8 accumulators of 16x16,
// K-loop of 32 steps of v_wmma_f32_16x16x4_f32. RELU folds the previous
// layer's activation into the A-fragment load (saves a full elementwise pass).
template<bool RELU>
__global__ __launch_bounds__(256) void k_gemm(const float* __restrict__ A,
                                              const float* __restrict__ W,
                                              float* __restrict__ C) {
    __shared__ float sW[DH * DH];
    int t = threadIdx.x;
    for (int i = t; i < DH * DH / 4; i += 256)
        ((float4*)sW)[i] = ((const float4*)W)[i];
    __syncthreads();

    int lane = t & 31;
    int wave = t >> 5;
    int half = lane >> 4;     // 0: lanes 0-15, 1: lanes 16-31
    int lm   = lane & 15;
    int row0 = blockIdx.x * 128 + wave * 16;
    if (row0 >= N_NODES) return;   // N_NODES % 16 == 0 -> no partial tiles

    v8f acc[8] = {};

    const float* arow = A + (size_t)(row0 + lm) * DH;
    #pragma unroll 2
    for (int k0 = 0; k0 < DH; k0 += 4) {
        // A fragment, 16x4 f32 layout: lane-half 0 holds K={k0,k0+1}, half 1 K={k0+2,k0+3}
        float2 av = *(const float2*)(arow + k0 + 2 * half);
        if (RELU) {
            av.x = fmaxf(av.x, 0.0f);
            av.y = fmaxf(av.y, 0.0f);
        }
        v2f a; a.x = av.x; a.y = av.y;
        #pragma unroll
        for (int nt = 0; nt < 8; nt++) {
            // B fragment, 4x16: row striped across lanes; halves hold K rows {0,1}/{2,3}
            v2f b;
            b.x = sW[(k0 + 2 * half)     * DH + nt * 16 + lm];
            b.y = sW[(k0 + 2 * half + 1) * DH + nt * 16 + lm];
            acc[nt] = __builtin_amdgcn_wmma_f32_16x16x4_f32(
                false, a, false, b, (short)0, acc[nt], false, false);
        }
    }

    // C/D layout: VGPR r holds M = r (lanes 0-15) / M = r+8 (lanes 16-31), N = lm
    #pragma unroll
    for (int nt = 0; nt < 8; nt++) {
        #pragma unroll
        for (int r = 0; r < 8; r++) {
            C[(size_t)(row0 + r + 8 * half) * DH + nt * 16 + lm] = acc[nt][r];
        }
    }
}

// ---------------- self-loop term + bias: A = B * dinv^2 + b -----------------
__global__ __launch_bounds__(256) void k_selfinit(const float* __restrict__ B,
                                                  const float* __restrict__ dinv,
                                                  const float* __restrict__ bias,
                                                  float* __restrict__ A) {
    int i = blockIdx.x * 256 + threadIdx.x;   // over N*32 (float4 per thread)
    if (i >= N_NODES * 32) return;
    int n  = i >> 5;
    int d4 = (i & 31) * 4;
    float di = dinv[n];
    float s  = di * di;
    float4 hv = *(const float4*)(B + (size_t)n * DH + d4);
    float4 bv = *(const float4*)(bias + d4);
    float4 o;
    o.x = hv.x * s + bv.x;
    o.y = hv.y * s + bv.y;
    o.z = hv.z * s + bv.z;
    o.w = hv.w * s + bv.w;
    *(float4*)(A + (size_t)n * DH + d4) = o;
}

// ---------------- edge scatter: A[dst] += B[src] * dinv[src]*dinv[dst] ------
// one wave per edge; each lane handles 4 contiguous dims (coalesced float4 gather)
__global__ __launch_bounds__(256) void k_edges(const int* __restrict__ src,
                                               const int* __restrict__ dst,
                                               const float* __restrict__ B,
                                               const float* __restrict__ dinv,
                                               float* __restrict__ A) {
    int gid = blockIdx.x * 256 + threadIdx.x;
    int e = gid >> 5;
    if (e >= N_EDGES) return;
    int lane = gid & 31;
    int s = src[e];
    int d = dst[e];
    float norm = dinv[s] * dinv[d];
    int d4 = lane * 4;
    float4 hv = *(const float4*)(B + (size_t)s * DH + d4);
    float* out = A + (size_t)d * DH + d4;
    atomicAdd(out + 0, hv.x * norm);
    atomicAdd(out + 1, hv.y * norm);
    atomicAdd(out + 2, hv.z * norm);
    atomicAdd(out + 3, hv.w * norm);
}

// ---------------- pooling sums (relu of last layer fused in) ----------------
__global__ __launch_bounds__(256) void k_pool(const float* __restrict__ A,
                                              const int* __restrict__ batch,
                                              float* __restrict__ sums) {
    int i = blockIdx.x * 256 + threadIdx.x;   // over N*32
    if (i >= N_NODES * 32) return;
    int n  = i >> 5;
    int d4 = (i & 31) * 4;
    int g  = batch[n];
    float4 hv = *(const float4*)(A + (size_t)n * DH + d4);
    hv.x = fmaxf(hv.x, 0.0f);
    hv.y = fmaxf(hv.y, 0.0f);
    hv.z = fmaxf(hv.z, 0.0f);
    hv.w = fmaxf(hv.w, 0.0f);
    float* out = sums + (size_t)g * DH + d4;
    atomicAdd(out + 0, hv.x);
    atomicAdd(out + 1, hv.y);
    atomicAdd(out + 2, hv.z);
    atomicAdd(out + 3, hv.w);
}

// ---------------- final linear: out[g] = (sums[g,:]/cnt) . lin_w + lin_b ----
__global__ __launch_bounds__(256) void k_final(const float* __restrict__ sums,
                                               const float* __restrict__ counts,
                                               const float* __restrict__ lin_w,
                                               const float* __restrict__ lin_b,
                                               float* __restrict__ out) {
    int g = blockIdx.x * 256 + threadIdx.x;
    if (g >= N_GRAPHS) return;
    float c = fmaxf(counts[g], 1.0f);
    float acc = 0.0f;
    #pragma unroll 4
    for (int d = 0; d < DH; d++) acc += sums[(size_t)g * DH + d] * lin_w[d];
    out[g] = acc / c + lin_b[0];
}

extern "C" void kernel_launch(void* const* d_in, const int* in_sizes, int n_in,
                              void* d_out, int out_size, void* d_ws, size_t ws_size,
                              hipStream_t stream) {
    const float* x     = (const float*)d_in[0];
    const int*   edge  = (const int*)d_in[1];
    const int*   src   = edge;             // edge_index[0, :]
    const int*   dst   = edge + N_EDGES;   // edge_index[1, :]
    const int*   batch = (const int*)d_in[2];
    const float* w0    = (const float*)d_in[3];
    const float* b0    = (const float*)d_in[4];
    const float* wsL   = (const float*)d_in[5];
    const float* bsL   = (const float*)d_in[6];
    const float* lin_w = (const float*)d_in[7];
    const float* lin_b = (const float*)d_in[8];
    float*       out   = (float*)d_out;

    // workspace: A | B | dinv | sums | counts  (~103 MB; A+B stay L2-resident)
    float* bufA   = (float*)d_ws;
    float* bufB   = bufA + (size_t)N_NODES * DH;
    float* dinv   = bufB + (size_t)N_NODES * DH;
    float* sums   = dinv + N_NODES;
    float* counts = sums + (size_t)N_GRAPHS * DH;

    const int blk256_N   = (N_NODES + 255) / 256;
    const int blk256_E   = (N_EDGES + 255) / 256;
    const int blk_gemm   = (N_NODES + 127) / 128;
    const int blk_nd32   = (N_NODES * 32 + 255) / 256;
    const int blk_edge32 = (int)(((size_t)N_EDGES * 32 + 255) / 256);

    k_init  <<<blk256_N, 256, 0, stream>>>(dinv, sums, counts);
    k_deg   <<<blk256_E, 256, 0, stream>>>(dst, dinv);
    k_dinv  <<<blk256_N, 256, 0, stream>>>(dinv);
    k_counts<<<blk256_N, 256, 0, stream>>>(batch, counts);

    const float* in = x;
    for (int l = 0; l < N_LAYERS; l++) {
        const float* W = (l == 0) ? w0 : wsL + (size_t)(l - 1) * DH * DH;
        const float* b = (l == 0) ? b0 : bsL + (size_t)(l - 1) * DH;
        if (l == 0)
            k_gemm<false><<<blk_gemm, 256, 0, stream>>>(in, W, bufB);
        else
            k_gemm<true> <<<blk_gemm, 256, 0, stream>>>(in, W, bufB);
        k_selfinit<<<blk_nd32,   256, 0, stream>>>(bufB, dinv, b, bufA);
        k_edges   <<<blk_edge32, 256, 0, stream>>>(src, dst, bufB, dinv, bufA);
        in = bufA;
    }

    k_pool <<<blk_nd32, 256, 0, stream>>>(bufA, batch, sums);
    k_final<<<(N_GRAPHS + 255) / 256, 256, 0, stream>>>(sums, counts, lin_w, lin_b, out);
}